// LambdaLoss_48000554500846
// MI455X (gfx1250) — compile-verified
//
#include <hip/hip_runtime.h>
#include <hip/hip_bf16.h>

// LambdaLoss (NDCG-weighted pairwise) for B=32, N=2048, f32.
// Phase 1: per-row bitonic sort (descending by preds, targets as payload) in LDS.
// Phase 2: tiled N x N pairwise sum; per-pair value fed to the matrix pipe via
//          V_WMMA_F32_16X16X4_F32 with A = ones(16x4) as a 64-wide fused add.
//          Sigmoid via gfx1250 v_tanh_f32 (1 TRANS op/pair) when available,
//          else v_exp_f32 + v_rcp_f32.
// Phase 3: deterministic ordered reduction of per-block partials.

#define BATCH       32
#define NITEMS      2048
#define SORT_THREADS 1024
#define P2_THREADS  256
#define P2_IBLK     256
#define JCHUNK      256

typedef __attribute__((ext_vector_type(2))) float v2f;
typedef __attribute__((ext_vector_type(8))) float v8f;

// sigmoid(-x) where hx = x/2:  0.5 - 0.5*tanh(x/2)  == 1/(1+exp(x))
#if __has_builtin(__builtin_amdgcn_tanhf)
  #define LL_HAVE_TANH 1
  __device__ __forceinline__ float sigmoid_of_neg_half(float hx) {
      return __builtin_fmaf(-0.5f, __builtin_amdgcn_tanhf(hx), 0.5f);
  }
#elif __has_builtin(__builtin_amdgcn_tanh_f32)
  #define LL_HAVE_TANH 1
  __device__ __forceinline__ float sigmoid_of_neg_half(float hx) {
      return __builtin_fmaf(-0.5f, __builtin_amdgcn_tanh_f32(hx), 0.5f);
  }
#else
  // Fallback: rcp(1 + exp2(2*log2(e) * hx)); raw v_rcp_f32 (~1 ulp).
  __device__ __forceinline__ float sigmoid_of_neg_half(float hx) {
      return __builtin_amdgcn_rcpf(
          1.0f + __builtin_amdgcn_exp2f(hx * 2.8853900817779268f));
  }
#endif

__global__ __launch_bounds__(SORT_THREADS)
void LambdaLoss_sort_rows(const float* __restrict__ preds,
                          const float* __restrict__ targets,
                          float* __restrict__ sp,
                          float* __restrict__ st) {
    __shared__ float key[NITEMS];
    __shared__ float val[NITEMS];
    const int b   = blockIdx.x;
    const int tid = threadIdx.x;
    const float* pr = preds   + b * NITEMS;
    const float* tg = targets + b * NITEMS;

    for (int t = tid; t < NITEMS; t += SORT_THREADS) {
        key[t] = pr[t];
        val[t] = tg[t];
    }

    // Bitonic sort, descending by key. (__syncthreads at top covers the load.)
    for (unsigned k = 2; k <= NITEMS; k <<= 1) {
        for (unsigned j = k >> 1; j > 0; j >>= 1) {
            __syncthreads();
            for (unsigned t = tid; t < NITEMS; t += SORT_THREADS) {
                unsigned ixj = t ^ j;
                if (ixj > t) {
                    float a = key[t], c = key[ixj];
                    bool descSeg = ((t & k) == 0);
                    bool doSwap  = descSeg ? (a < c) : (a > c);
                    if (doSwap) {
                        key[t] = c; key[ixj] = a;
                        float va = val[t]; val[t] = val[ixj]; val[ixj] = va;
                    }
                }
            }
        }
    }
    __syncthreads();

    for (int t = tid; t < NITEMS; t += SORT_THREADS) {
        sp[b * NITEMS + t] = key[t];
        st[b * NITEMS + t] = val[t];
    }
}

__global__ __launch_bounds__(P2_THREADS)
void LambdaLoss_pairwise_wmma(const float* __restrict__ sp,
                              const float* __restrict__ st,
                              float* __restrict__ partials) {
    __shared__ float hpj[JCHUNK];   // 0.5 * sorted pred (half-scores)
    __shared__ float stj[JCHUNK];
    __shared__ float dj [JCHUNK];
    __shared__ float red[P2_THREADS];

    const int blocksPerRow = NITEMS / P2_IBLK;      // 8
    const int b   = blockIdx.x / blocksPerRow;
    const int ib  = blockIdx.x % blocksPerRow;
    const int tid = threadIdx.x;
    const int i   = ib * P2_IBLK + tid;             // this thread's rank i

    const float* rowp = sp + b * NITEMS;
    const float* rowt = st + b * NITEMS;

    const float hpi = 0.5f * rowp[i];               // half-score: tanh arg = hpi-hpj
    const float sti = rowt[i];
    const float di  = __log2f((float)(i + 3));      // disc_i = log2(rank_i + 3)

    v8f acc = {0.f, 0.f, 0.f, 0.f, 0.f, 0.f, 0.f, 0.f};
    const v2f onesA = {1.0f, 1.0f};                 // A = ones(16x4): row-sum operator

    for (int jc = 0; jc < NITEMS; jc += JCHUNK) {
        __syncthreads();
        hpj[tid] = 0.5f * rowp[jc + tid];
        stj[tid] = rowt[jc + tid];
        dj [tid] = __log2f((float)(jc + tid + 3));
        __syncthreads();

        // Prefetch next chunk toward the WGP while this one is consumed
        // (lowers to global_prefetch_b8; data is L2-resident regardless).
        if (jc + JCHUNK < NITEMS) {
            __builtin_prefetch(rowp + jc + JCHUNK + tid, 0, 3);
            __builtin_prefetch(rowt + jc + JCHUNK + tid, 0, 3);
        }

        // No divergence here: EXEC is all-ones for every WMMA.
        #pragma unroll 8
        for (int jj = 0; jj < JCHUNK; jj += 2) {
            float s0 = sigmoid_of_neg_half(hpi - hpj[jj]);
            float v0 = s0 * fabsf((sti - stj[jj]) * (dj[jj] - di));
            float s1 = sigmoid_of_neg_half(hpi - hpj[jj + 1]);
            float v1 = s1 * fabsf((sti - stj[jj + 1]) * (dj[jj + 1] - di));
            v2f bv = {v0, v1};
            // D = ones(16x4) x B(4x16) + C : accumulates 64 pair values/instr.
            acc = __builtin_amdgcn_wmma_f32_16x16x4_f32(
                false, onesA, false, bv, (short)0, acc, false, false);
        }
    }

    // Each D column sum is replicated across 16 rows; summing every acc element
    // over all lanes yields 16x the wave's true sum (corrected in final kernel).
    float p = acc[0] + acc[1] + acc[2] + acc[3] +
              acc[4] + acc[5] + acc[6] + acc[7];
    red[tid] = p;
    __syncthreads();
    if (tid == 0) {
        float s = 0.f;
        for (int t = 0; t < P2_THREADS; ++t) s += red[t];  // fixed order
        partials[blockIdx.x] = s;
    }
}

__global__ void LambdaLoss_final_reduce(const float* __restrict__ partials,
                                        float* __restrict__ out, int n) {
    if (threadIdx.x == 0 && blockIdx.x == 0) {
        float s = 0.f;
        for (int t = 0; t < n; ++t) s += partials[t];      // deterministic order
        out[0] = s * (1.0f / (16.0f * (float)BATCH));      // /16 WMMA replication, /B
    }
}

extern "C" void kernel_launch(void* const* d_in, const int* in_sizes, int n_in,
                              void* d_out, int out_size, void* d_ws, size_t ws_size,
                              hipStream_t stream) {
    (void)in_sizes; (void)n_in; (void)out_size; (void)ws_size;
    const float* preds   = (const float*)d_in[0];
    const float* targets = (const float*)d_in[1];
    float* out = (float*)d_out;

    float* sp       = (float*)d_ws;                 // [32][2048] sorted preds
    float* st       = sp + BATCH * NITEMS;          // [32][2048] sorted targets
    float* partials = st + BATCH * NITEMS;          // [256] block partials

    const int nblk2 = BATCH * (NITEMS / P2_IBLK);   // 256 blocks

    LambdaLoss_sort_rows<<<BATCH, SORT_THREADS, 0, stream>>>(preds, targets, sp, st);
    LambdaLoss_pairwise_wmma<<<nblk2, P2_THREADS, 0, stream>>>(sp, st, partials);
    LambdaLoss_final_reduce<<<1, 32, 0, stream>>>(partials, out, nblk2);
}